// ResGraphModule_64622077935967
// MI455X (gfx1250) — compile-verified
//
#include <hip/hip_runtime.h>
#include <hip/hip_bf16.h>

// ---------------------------------------------------------------------------
// ResGraphModule on MI455X (gfx1250, wave32).
// Bandwidth-bound (~1.2 GB moved vs 12.3 GFLOP) -> keep fp32 exact and use
// V_WMMA_F32_16X16X4_F32 for every GEMM. 16x16 tiles, LDS staging for the
// edge-message transpose, hardware f32 atomics for segment_sum.
// ---------------------------------------------------------------------------

#define NN 100000
#define NE 1600000
#define C  64
#define EC 32

typedef __attribute__((ext_vector_type(2))) float v2f;
typedef __attribute__((ext_vector_type(4))) float v4f;
typedef __attribute__((ext_vector_type(8))) float v8f;

static __device__ __forceinline__ v8f wmma4(v2f a, v2f b, v8f c) {
  // D = A(16x4,f32) * B(4x16,f32) + C(16x16,f32)
  return __builtin_amdgcn_wmma_f32_16x16x4_f32(false, a, false, b,
                                               (short)0, c, false, false);
}

static __device__ __forceinline__ v8f vzero8() {
  v8f z = {0.f, 0.f, 0.f, 0.f, 0.f, 0.f, 0.f, 0.f};
  return z;
}

// ---------------------------------------------------------------------------
// Kernel 0: zero the aggregation buffer (atomically accumulated each call)
// ---------------------------------------------------------------------------
__global__ void k_zero(float* __restrict__ p, int n) {
  int i = blockIdx.x * blockDim.x + threadIdx.x;
  if (i < n) p[i] = 0.f;
}

// ---------------------------------------------------------------------------
// Kernel 1: m = x @ W   ([N,64] x [64,64])
// One wave per 16x16 output tile; blockDim=128 -> 4 col tiles per row tile.
// ---------------------------------------------------------------------------
__global__ void k_node_lin(const float* __restrict__ x,
                           const float* __restrict__ W,
                           float* __restrict__ m) {
  const int lane = threadIdx.x & 31;
  const int wave = threadIdx.x >> 5;      // col tile 0..3
  const int rt   = blockIdx.x;            // row tile
  const int half = lane >> 4;             // K sub-slot (A/B frag layout)
  const int mrow = (lane & 15);
  const int r    = rt * 16 + mrow;
  const int col  = wave * 16 + mrow;

  v8f acc = vzero8();
#pragma unroll
  for (int k = 0; k < C; k += 4) {
    const int kb = k + half * 2;
    v2f a; a.x = x[r * C + kb];        a.y = x[r * C + kb + 1];
    v2f b; b.x = W[kb * C + col];      b.y = W[(kb + 1) * C + col];
    acc = wmma4(a, b, acc);
  }
#pragma unroll
  for (int v = 0; v < 8; ++v) {
    const int ro = rt * 16 + v + half * 8;
    m[ro * C + col] = acc[v];
  }
}

// ---------------------------------------------------------------------------
// Kernel 2: per 16-edge tile:  ew = edge_attr[16,32] @ We^T[32,64]  (WMMA),
// stage ew in LDS, then msg = m[src]*ew and atomic scatter-add into agg[dst].
// blockDim=128 (4 waves, each owns one 16-edge tile + its LDS slice).
// ---------------------------------------------------------------------------
__global__ void k_edge(const float* __restrict__ ea,
                       const float* __restrict__ We,
                       const long long* __restrict__ ei,
                       const float* __restrict__ m,
                       float* __restrict__ agg) {
  __shared__ float ew_s[4][16][C + 4];    // +4 pad: conflict-free row reads

  const int lane = threadIdx.x & 31;
  const int wave = threadIdx.x >> 5;
  const int e0   = (blockIdx.x * 4 + wave) * 16;
  const int half = lane >> 4;
  const int mrow = (lane & 15);
  const int erow = e0 + mrow;

  v8f acc[4];
#pragma unroll
  for (int t = 0; t < 4; ++t) acc[t] = vzero8();

#pragma unroll
  for (int k = 0; k < EC; k += 4) {
    const int kb = k + half * 2;
    v2f a; a.x = ea[erow * EC + kb]; a.y = ea[erow * EC + kb + 1];
#pragma unroll
    for (int t = 0; t < 4; ++t) {
      const int col = t * 16 + mrow;     // output channel j; B[k][j]=We[j*EC+k]
      v2f b; b.x = We[col * EC + kb]; b.y = We[col * EC + kb + 1];
      acc[t] = wmma4(a, b, acc[t]);
    }
  }

  // D layout -> LDS row-major ew[edge][channel]
#pragma unroll
  for (int t = 0; t < 4; ++t) {
#pragma unroll
    for (int v = 0; v < 8; ++v) {
      ew_s[wave][v + half * 8][t * 16 + mrow] = acc[t][v];
    }
  }
  __syncthreads();

  // Phase 2: 2 lanes per edge, 32 contiguous channels each.
  const int el       = lane >> 1;
  const long long e  = (long long)e0 + el;
  const int cb       = (lane & 1) * 32;
  const long long s  = ei[e];
  const long long d  = ei[(long long)NE + e];
  const float* mr    = m + s * C + cb;
  float* ar          = agg + d * C + cb;

#pragma unroll
  for (int i = 0; i < 8; ++i) {
    v4f mv  = *(const v4f*)(mr + i * 4);
    v4f wv  = *(const v4f*)(&ew_s[wave][el][cb + i * 4]);
    v4f val = mv * wv;
#pragma unroll
    for (int j = 0; j < 4; ++j) {
      unsafeAtomicAdd(ar + i * 4 + j, val[j]);   // global_atomic_add_f32
    }
  }
}

// ---------------------------------------------------------------------------
// Kernel 3: fused GRU + relu + residual.
// Per 16-node tile: wave w owns cols [16w,16w+16) of all 3 gates.
// gi = agg @ W_ih^T, gh = x @ W_hh^T computed in registers (6 WMMA accums),
// then gate math + 0.5*relu(h)+0.5*x stored straight to d_out.
// ---------------------------------------------------------------------------
__global__ void k_gru(const float* __restrict__ x,
                      const float* __restrict__ agg,
                      const float* __restrict__ Wih,
                      const float* __restrict__ Whh,
                      const float* __restrict__ bih,
                      const float* __restrict__ bhh,
                      float* __restrict__ out) {
  const int lane = threadIdx.x & 31;
  const int wave = threadIdx.x >> 5;      // 16-col slice within each gate
  const int nt   = blockIdx.x;
  const int half = lane >> 4;
  const int mrow = (lane & 15);
  const int r    = nt * 16 + mrow;
  const int cl   = wave * 16 + mrow;      // 0..63: column within a gate

  v8f gi[3], gh[3];
#pragma unroll
  for (int g = 0; g < 3; ++g) { gi[g] = vzero8(); gh[g] = vzero8(); }

#pragma unroll
  for (int k = 0; k < C; k += 4) {
    const int kb = k + half * 2;
    v2f aa; aa.x = agg[r * C + kb]; aa.y = agg[r * C + kb + 1];
    v2f ax; ax.x = x[r * C + kb];   ax.y = x[r * C + kb + 1];
#pragma unroll
    for (int g = 0; g < 3; ++g) {
      const int jr = g * C + cl;          // output row of W_*: B[k][j]=W[j*C+k]
      v2f bi; bi.x = Wih[jr * C + kb]; bi.y = Wih[jr * C + kb + 1];
      gi[g] = wmma4(aa, bi, gi[g]);
      v2f bh; bh.x = Whh[jr * C + kb]; bh.y = Whh[jr * C + kb + 1];
      gh[g] = wmma4(ax, bh, gh[g]);
    }
  }

  const float bri = bih[0 * C + cl], bzi = bih[1 * C + cl], bni = bih[2 * C + cl];
  const float brh = bhh[0 * C + cl], bzh = bhh[1 * C + cl], bnh = bhh[2 * C + cl];

#pragma unroll
  for (int v = 0; v < 8; ++v) {
    const int ro = nt * 16 + v + half * 8;
    const float xv = x[ro * C + cl];
    const float rg = 1.f / (1.f + __expf(-(gi[0][v] + bri + gh[0][v] + brh)));
    const float zg = 1.f / (1.f + __expf(-(gi[1][v] + bzi + gh[1][v] + bzh)));
    const float ng = tanhf(gi[2][v] + bni + rg * (gh[2][v] + bnh));
    const float h  = (1.f - zg) * ng + zg * xv;
    const float o  = fmaxf(h, 0.f);
    out[ro * C + cl] = 0.5f * o + 0.5f * xv;
  }
}

// ---------------------------------------------------------------------------
extern "C" void kernel_launch(void* const* d_in, const int* in_sizes, int n_in,
                              void* d_out, int out_size, void* d_ws, size_t ws_size,
                              hipStream_t stream) {
  const float*     x   = (const float*)d_in[0];
  const long long* ei  = (const long long*)d_in[1];   // int64 [2, NE]
  const float*     ea  = (const float*)d_in[2];
  const float*     We  = (const float*)d_in[3];       // [C, EC]
  const float*     W   = (const float*)d_in[4];       // [C, C]
  const float*     Wih = (const float*)d_in[5];       // [3C, C]
  const float*     Whh = (const float*)d_in[6];       // [3C, C]
  const float*     bih = (const float*)d_in[7];
  const float*     bhh = (const float*)d_in[8];
  float*           out = (float*)d_out;

  float* m   = (float*)d_ws;                          // [NN, C]
  float* agg = m + (size_t)NN * C;                    // [NN, C]

  const int nAgg = NN * C;
  k_zero<<<(nAgg + 255) / 256, 256, 0, stream>>>(agg, nAgg);
  k_node_lin<<<NN / 16, 128, 0, stream>>>(x, W, m);
  k_edge<<<NE / 64, 128, 0, stream>>>(ea, We, ei, m, agg);
  k_gru<<<NN / 16, 128, 0, stream>>>(x, agg, Wih, Whh, bih, bhh, out);
}